// GCN_3676492005492
// MI455X (gfx1250) — compile-verified
//
#include <hip/hip_runtime.h>
#include <hip/hip_bf16.h>

#define N_NODES  50000
#define N_EDGES  800000
#define N_GRAPHS 128
#define F_IN     256
#define F_MID    128
#define F_OUT    32

typedef __attribute__((ext_vector_type(2))) float v2f;
typedef __attribute__((ext_vector_type(8))) float v8f;

// ---------------------------------------------------------------------------
// Degree / normalization
// ---------------------------------------------------------------------------
__global__ void fill_deg_kernel(float* deg) {
    int i = blockIdx.x * blockDim.x + threadIdx.x;
    if (i < N_NODES) deg[i] = 1.0f;  // self-loop
}

__global__ void count_deg_kernel(const int* __restrict__ row, float* deg) {
    int e = blockIdx.x * blockDim.x + threadIdx.x;
    if (e < N_EDGES) atomicAdd(&deg[row[e]], 1.0f);
}

__global__ void dinv_kernel(float* deg) {
    int i = blockIdx.x * blockDim.x + threadIdx.x;
    if (i < N_NODES) {
        float d = deg[i];
        deg[i] = (d > 0.0f) ? rsqrtf(d) : 0.0f;
    }
}

// ---------------------------------------------------------------------------
// WMMA fp32 GEMM:  D[M,N] = A[M,K] @ B[K,N]   (optionally relu(A))
// One wave computes NT adjacent 16x16 tiles (reusing the A fragment) with
// V_WMMA_F32_16X16X4_F32.  A tile (16 x K) staged in LDS with +4-float row
// padding (conflict-free ds_load_b64 frag reads, 16B-aligned rows).  N and K
// are compile-time so all B loads / D stores use immediate offsets off a
// single base address (no per-iteration 64-bit address math).
// ---------------------------------------------------------------------------
template <int K, int N, int NW, int NT, bool RELU>
__global__ void gemm_wmma_kernel(const float* __restrict__ A,
                                 const float* __restrict__ B,
                                 float* __restrict__ D) {
    __shared__ float As[16][K + 4];

    const int tid  = threadIdx.x;
    const int wave = tid >> 5;
    const int lane = tid & 31;
    const int idx  = lane & 15;       // position within 16-lane half
    const int half = lane >> 4;       // 0 or 1
    const int m0   = blockIdx.x * 16;
    const int n0   = (blockIdx.y * NW + wave) * (16 * NT);

    // Cooperative staged load of the 16 x K A-tile (float4 global + LDS stores)
    const int nvec = (16 * K) / 4;
    for (int v = tid; v < nvec; v += 32 * NW) {
        int r = v / (K / 4);
        int q = v % (K / 4);
        float4 a4 = *reinterpret_cast<const float4*>(A + (size_t)(m0 + r) * K + q * 4);
        if (RELU) {
            a4.x = fmaxf(a4.x, 0.0f); a4.y = fmaxf(a4.y, 0.0f);
            a4.z = fmaxf(a4.z, 0.0f); a4.w = fmaxf(a4.w, 0.0f);
        }
        *reinterpret_cast<float4*>(&As[r][q * 4]) = a4;
    }
    __syncthreads();

    // Fixed per-lane base pointers; all subsequent offsets are constants.
    const float* Bbase = B + n0 + idx;
    float*       Dbase = D + (size_t)(m0 + 8 * half) * N + n0 + idx;

    v8f acc[NT];
#pragma unroll
    for (int t = 0; t < NT; ++t) acc[t] = (v8f){};

#pragma unroll 8
    for (int k0 = 0; k0 < K; k0 += 4) {
        // A frag: lane=M (idx), VGPR0/1 = K {k0,k0+1} (half 0) / {k0+2,k0+3} (half 1)
        v2f a = *reinterpret_cast<const v2f*>(&As[idx][k0 + 2 * half]);
#pragma unroll
        for (int t = 0; t < NT; ++t) {
            // B frag: lane=N (idx), VGPR0/1 = rows k0+2*half, k0+2*half+1
            v2f b;
            b.x = Bbase[(k0 + 2 * half) * N + t * 16];
            b.y = Bbase[(k0 + 2 * half + 1) * N + t * 16];
            acc[t] = __builtin_amdgcn_wmma_f32_16x16x4_f32(
                /*neg_a=*/false, a, /*neg_b=*/false, b,
                /*c_mod=*/(short)0, acc[t], /*reuse_a=*/false, /*reuse_b=*/false);
        }
    }

    // D layout: VGPR v -> M = m0 + v + 8*half, N = n0 + t*16 + idx
#pragma unroll
    for (int t = 0; t < NT; ++t)
#pragma unroll
        for (int v = 0; v < 8; ++v)
            Dbase[(size_t)v * N + t * 16] = acc[t][v];
}

// ---------------------------------------------------------------------------
// out[i,f] = h[i,f]*dinv[i]^2 + b[f]     (self-loop contribution + bias)
// float4-vectorized; F multiple of 4.
// ---------------------------------------------------------------------------
__global__ void init_agg_kernel(const float* __restrict__ h,
                                const float* __restrict__ dinv,
                                const float* __restrict__ bias,
                                float* __restrict__ out, int F, int total4) {
    int t = blockIdx.x * blockDim.x + threadIdx.x;
    if (t >= total4) return;
    int base = t * 4;
    int i = base / F;
    int f = base % F;
    float w = dinv[i] * dinv[i];
    float4 hv = *reinterpret_cast<const float4*>(h + base);
    float4 bv = *reinterpret_cast<const float4*>(bias + f);
    float4 o;
    o.x = hv.x * w + bv.x; o.y = hv.y * w + bv.y;
    o.z = hv.z * w + bv.z; o.w = hv.w * w + bv.w;
    *reinterpret_cast<float4*>(out + base) = o;
}

// ---------------------------------------------------------------------------
// Edge scatter, F=128: one wave per edge, 4 feats per lane (float4 gather,
// 4x global_atomic_add_f32 scatter)
// ---------------------------------------------------------------------------
__global__ void scatter_f128_kernel(const int* __restrict__ row,
                                    const int* __restrict__ col,
                                    const float* __restrict__ dinv,
                                    const float* __restrict__ h,
                                    float* __restrict__ out) {
    int gw = (blockIdx.x * blockDim.x + threadIdx.x) >> 5;
    int lane = threadIdx.x & 31;
    if (gw >= N_EDGES) return;
    int r = row[gw], c = col[gw];
    float w = dinv[r] * dinv[c];
    float4 hv = *reinterpret_cast<const float4*>(h + (size_t)c * 128 + lane * 4);
    float* o = out + (size_t)r * 128 + lane * 4;
    atomicAdd(o + 0, hv.x * w);
    atomicAdd(o + 1, hv.y * w);
    atomicAdd(o + 2, hv.z * w);
    atomicAdd(o + 3, hv.w * w);
}

// Edge scatter, F=32: one wave per edge, 1 feat per lane
__global__ void scatter_f32_kernel(const int* __restrict__ row,
                                   const int* __restrict__ col,
                                   const float* __restrict__ dinv,
                                   const float* __restrict__ h,
                                   float* __restrict__ out) {
    int gw = (blockIdx.x * blockDim.x + threadIdx.x) >> 5;
    int lane = threadIdx.x & 31;
    if (gw >= N_EDGES) return;
    int r = row[gw], c = col[gw];
    float w = dinv[r] * dinv[c];
    float v = h[(size_t)c * 32 + lane] * w;
    atomicAdd(out + (size_t)r * 32 + lane, v);
}

// ---------------------------------------------------------------------------
// Pooling + log_softmax
// ---------------------------------------------------------------------------
__global__ void zero_pool_kernel(float* sums, float* cnt) {
    int i = blockIdx.x * blockDim.x + threadIdx.x;
    if (i < N_GRAPHS * F_OUT) sums[i] = 0.0f;
    if (i < N_GRAPHS) cnt[i] = 0.0f;
}

__global__ void pool_scatter_kernel(const int* __restrict__ batch,
                                    const float* __restrict__ h,
                                    float* __restrict__ sums,
                                    float* __restrict__ cnt) {
    int gw = (blockIdx.x * blockDim.x + threadIdx.x) >> 5;  // node id
    int lane = threadIdx.x & 31;
    if (gw >= N_NODES) return;
    int g = batch[gw];
    atomicAdd(&sums[(size_t)g * F_OUT + lane], h[(size_t)gw * F_OUT + lane]);
    if (lane == 0) atomicAdd(&cnt[g], 1.0f);
}

__global__ void logsoftmax_kernel(const float* __restrict__ sums,
                                  const float* __restrict__ cnt,
                                  float* __restrict__ out) {
    int g = blockIdx.x;
    int lane = threadIdx.x;
    float v = sums[(size_t)g * F_OUT + lane] / fmaxf(cnt[g], 1.0f);
    // wave32 max-reduce
    float m = v;
    for (int off = 16; off > 0; off >>= 1)
        m = fmaxf(m, __shfl_xor(m, off, 32));
    float e = expf(v - m);
    float s = e;
    for (int off = 16; off > 0; off >>= 1)
        s += __shfl_xor(s, off, 32);
    out[(size_t)g * F_OUT + lane] = (v - m) - logf(s);
}

// ---------------------------------------------------------------------------
// Launch
// ---------------------------------------------------------------------------
extern "C" void kernel_launch(void* const* d_in, const int* in_sizes, int n_in,
                              void* d_out, int out_size, void* d_ws, size_t ws_size,
                              hipStream_t stream) {
    const float* x  = (const float*)d_in[0];             // [50000,256]
    const int*   ei = (const int*)d_in[1];               // [2,800000]
    const int*   batch = (const int*)d_in[2];            // [50000]
    const float* W1 = (const float*)d_in[3];             // [256,128]
    const float* b1 = (const float*)d_in[4];             // [128]
    const float* W2 = (const float*)d_in[5];             // [128,32]
    const float* b2 = (const float*)d_in[6];             // [32]
    float* out = (float*)d_out;                          // [128,32]

    const int* row = ei;
    const int* col = ei + N_EDGES;

    float* ws   = (float*)d_ws;
    float* dinv = ws;                                    // 50016
    float* h    = dinv + 50016;                          // 50000*128 (h1, reused for h2)
    float* agg  = h + (size_t)N_NODES * F_MID;           // 50000*128 (agg1, reused agg2)
    float* sums = agg + (size_t)N_NODES * F_MID;         // 128*32
    float* cnt  = sums + N_GRAPHS * F_OUT;               // 128
    (void)ws_size; (void)n_in; (void)in_sizes; (void)out_size;

    // normalization coefficients (shared by both layers)
    fill_deg_kernel<<<(N_NODES + 255) / 256, 256, 0, stream>>>(dinv);
    count_deg_kernel<<<(N_EDGES + 255) / 256, 256, 0, stream>>>(row, dinv);
    dinv_kernel<<<(N_NODES + 255) / 256, 256, 0, stream>>>(dinv);

    // ---- layer 1: h1 = x @ W1 ; agg1 = Ahat @ h1 + b1 ----
    // 4 waves x (16x32) tiles -> block covers full N=128; grid.y = 1
    gemm_wmma_kernel<F_IN, F_MID, 4, 2, false>
        <<<dim3(N_NODES / 16, 1), 32 * 4, 0, stream>>>(x, W1, h);
    init_agg_kernel<<<((N_NODES * F_MID / 4) + 255) / 256, 256, 0, stream>>>(
        h, dinv, b1, agg, F_MID, N_NODES * F_MID / 4);
    scatter_f128_kernel<<<(N_EDGES * 32 + 255) / 256, 256, 0, stream>>>(
        row, col, dinv, h, agg);

    // ---- layer 2: h2 = relu(agg1) @ W2 ; agg2 = Ahat @ h2 + b2 ----
    // 2 waves x (16x16) tiles -> block covers full N=32; grid.y = 1
    gemm_wmma_kernel<F_MID, F_OUT, 2, 1, true>
        <<<dim3(N_NODES / 16, 1), 32 * 2, 0, stream>>>(agg, W2, h);
    init_agg_kernel<<<((N_NODES * F_OUT / 4) + 255) / 256, 256, 0, stream>>>(
        h, dinv, b2, agg, F_OUT, N_NODES * F_OUT / 4);
    scatter_f32_kernel<<<(N_EDGES * 32 + 255) / 256, 256, 0, stream>>>(
        row, col, dinv, h, agg);

    // ---- mean pool + log_softmax ----
    zero_pool_kernel<<<(N_GRAPHS * F_OUT + 255) / 256, 256, 0, stream>>>(sums, cnt);
    pool_scatter_kernel<<<(N_NODES * 32 + 255) / 256, 256, 0, stream>>>(
        batch, agg, sums, cnt);
    logsoftmax_kernel<<<N_GRAPHS, F_OUT, 0, stream>>>(sums, cnt, out);
}